// GaussianSelfAttention_40810779247047
// MI455X (gfx1250) — compile-verified
//
#include <hip/hip_runtime.h>
#include <math.h>

typedef float v2f __attribute__((ext_vector_type(2)));
typedef float v8f __attribute__((ext_vector_type(8)));

#define MTOT 12608   // B*S = 64*197
#define DD   768
#define SEQ  197
#define PP   196

#define BPAIR_STRIDE 544   // 16 row-pairs, padded: 544 % 64 == 32 -> half-wave bank sets disjoint

// ---------------------------------------------------------------------------
// Kernel 1: fused QKV projection GEMM using native fp32 WMMA (16x16x4).
// Block tile: 64(M) x 256(N) x 32(K). 8 waves (2x4), each wave a 32x64 tile
// (2x4 accumulators of 16x16). B stored in LDS row-pair interleaved so each
// B fragment is one contiguous b64 per lane. Next global tile prefetched to
// registers while the current LDS tile is consumed.
// gridDim = (197, 3, 3); z selects {Wq,bq,q} / {Wk,bk,k} / {Wv,bv,v}.
// ---------------------------------------------------------------------------
__global__ __launch_bounds__(256) void qkv_gemm(
    const float* __restrict__ x,
    const float* __restrict__ Wq, const float* __restrict__ bq,
    const float* __restrict__ Wk, const float* __restrict__ bk,
    const float* __restrict__ Wv, const float* __restrict__ bv,
    float* __restrict__ ws)
{
    const int z = blockIdx.z;
    const float* __restrict__ W    = (z == 0) ? Wq : (z == 1) ? Wk : Wv;
    const float* __restrict__ bias = (z == 0) ? bq : (z == 1) ? bk : bv;
    float* __restrict__ out = ws + (size_t)z * MTOT * DD;

    // A tile 64x32, row stride 36 floats: b64 column reads are bank-conflict
    // free (36*l mod 64 cycles through 16 distinct even banks; half-waves use
    // disjoint bank classes mod 4). 16B-aligned float4 stores preserved.
    __shared__ __align__(16) float As[64 * 36];
    // B tile 32x256 stored as 16 row-pairs: element (k, n) lives at
    // Bs[(k>>1)*BPAIR_STRIDE + 2*n + (k&1)]  -> fragment = contiguous b64.
    __shared__ __align__(16) float Bs[16 * BPAIR_STRIDE];

    const int tid  = threadIdx.x;
    const int lane = tid & 31;
    const int wid  = tid >> 5;
    const int mW   = (wid >> 2) * 32;    // wave M origin: 0 or 32
    const int nW   = (wid & 3) * 64;     // wave N origin: 0..192
    const int lh   = lane >> 4;          // half-wave (K split)
    const int lm   = lane & 15;

    const int m0 = blockIdx.x * 64;
    const int n0 = blockIdx.y * 256;

    v8f acc[2][4];
#pragma unroll
    for (int mi = 0; mi < 2; ++mi)
#pragma unroll
        for (int ni = 0; ni < 4; ++ni) { v8f zz = {}; acc[mi][ni] = zz; }

    // Loader index precompute.
    int aRow[2], aCol[2];       // A: 64 rows x 8 float4/row, 2 per thread
#pragma unroll
    for (int it = 0; it < 2; ++it) {
        int id = tid + it * 256;
        aRow[it] = id >> 3;
        aCol[it] = (id & 7) * 4;
    }
    int bRp[4], bCol[4];        // B: 16 row-pairs x 64 float4/row, 4 pairs per thread
#pragma unroll
    for (int it = 0; it < 4; ++it) {
        int id = tid + it * 256;
        bRp[it]  = id >> 6;
        bCol[it] = (id & 63) * 4;
    }

    float4 aReg[2], bLo[4], bHi[4];

    auto load_tile = [&](int k0) {
#pragma unroll
        for (int it = 0; it < 2; ++it)
            aReg[it] = *(const float4*)(x + (size_t)(m0 + aRow[it]) * DD + k0 + aCol[it]);
#pragma unroll
        for (int it = 0; it < 4; ++it) {
            const float* base = W + (size_t)(k0 + 2 * bRp[it]) * DD + n0 + bCol[it];
            bLo[it] = *(const float4*)(base);
            bHi[it] = *(const float4*)(base + DD);
        }
    };

    load_tile(0);

    for (int k0 = 0;; k0 += 32) {
        __syncthreads();   // previous compute done before LDS overwrite
#pragma unroll
        for (int it = 0; it < 2; ++it)
            *(float4*)(&As[aRow[it] * 36 + aCol[it]]) = aReg[it];
#pragma unroll
        for (int it = 0; it < 4; ++it) {
            float* bb = &Bs[bRp[it] * BPAIR_STRIDE + bCol[it] * 2];
            *(float2*)(bb + 0) = make_float2(bLo[it].x, bHi[it].x);
            *(float2*)(bb + 2) = make_float2(bLo[it].y, bHi[it].y);
            *(float2*)(bb + 4) = make_float2(bLo[it].z, bHi[it].z);
            *(float2*)(bb + 6) = make_float2(bLo[it].w, bHi[it].w);
        }
        __syncthreads();

        const bool more = (k0 + 32) < DD;
        if (more) load_tile(k0 + 32);   // prefetch next tile during compute

#pragma unroll
        for (int kk = 0; kk < 32; kk += 4) {
            // A fragment (16x4): lanes 0-15 K=kk,kk+1; lanes 16-31 K=kk+2,kk+3
            v2f a[2], b[4];
            const int acol = kk + lh * 2;
#pragma unroll
            for (int mi = 0; mi < 2; ++mi)
                a[mi] = *(const v2f*)(&As[(mW + mi * 16 + lm) * 36 + acol]);
            // B fragment (4x16): VGPR0 = row brow, VGPR1 = row brow+1 -> one b64
            const int bpr = (kk >> 1) + lh;   // (kk + lh*2) / 2, brow even
#pragma unroll
            for (int ni = 0; ni < 4; ++ni)
                b[ni] = *(const v2f*)(&Bs[bpr * BPAIR_STRIDE + (nW + ni * 16 + lm) * 2]);
#pragma unroll
            for (int mi = 0; mi < 2; ++mi)
#pragma unroll
                for (int ni = 0; ni < 4; ++ni)
                    acc[mi][ni] = __builtin_amdgcn_wmma_f32_16x16x4_f32(
                        false, a[mi], false, b[ni],
                        (short)0, acc[mi][ni], false, false);
        }
        if (!more) break;
    }

    // Epilogue: bias add + store. C layout: VGPR j => row j (lanes 0-15) /
    // row j+8 (lanes 16-31); col = lane%16 within the 16x16 tile.
#pragma unroll
    for (int mi = 0; mi < 2; ++mi)
#pragma unroll
        for (int ni = 0; ni < 4; ++ni) {
            int col = n0 + nW + ni * 16 + lm;
            float bval = bias[col];
            int rbase = m0 + mW + mi * 16 + lh * 8;
#pragma unroll
            for (int j = 0; j < 8; ++j)
                out[(size_t)(rbase + j) * DD + col] = acc[mi][ni][j] + bval;
        }
}

// ---------------------------------------------------------------------------
// Kernel 2: bilinear-index gather + 4-way attention. One wave32 per token.
// ---------------------------------------------------------------------------
__device__ __forceinline__ int wrap197(int t) {
    int r = t % SEQ;
    return (r < 0) ? r + SEQ : r;   // Python floor-mod semantics
}

__global__ __launch_bounds__(256) void gauss_attn(
    const float* __restrict__ ws,
    const float* __restrict__ norm_x, const float* __restrict__ norm_y,
    const float* __restrict__ avgs,   const float* __restrict__ std_devs,
    const int*   __restrict__ img_ids,
    float* __restrict__ out)
{
    const size_t PLANE = (size_t)MTOT * DD;
    const float* __restrict__ q = ws;
    const float* __restrict__ k = ws + PLANE;
    const float* __restrict__ v = ws + 2 * PLANE;

    const int lane = threadIdx.x & 31;
    const int wid  = threadIdx.x >> 5;
    const int t    = blockIdx.x * 8 + wid;
    if (t >= MTOT) return;
    const int b = t / SEQ;
    const int s = t % SEQ;

    float* __restrict__ orow = out + (size_t)t * DD;

    if (s == 0) {
        // class token: keys/values are all-ones -> softmax uniform -> output = 1
        float4 one = make_float4(1.f, 1.f, 1.f, 1.f);
#pragma unroll
        for (int j = 0; j < 6; ++j)
            *(float4*)(orow + (lane + 32 * j) * 4) = one;
        return;
    }

    // All lanes compute the 4 bilinear indices identically.
    const int p   = s - 1;
    const int img = img_ids[b];
    const size_t a0 = ((size_t)img * 2 + 0) * PP + p;
    const size_t a1 = ((size_t)img * 2 + 1) * PP + p;
    float kx = (norm_x[b * PP + p] - avgs[a0]) / std_devs[a0];
    float ky = (norm_y[b * PP + p] - avgs[a1]) / std_devs[a1];
    float kxc = ceilf(kx), kxf = floorf(kx);
    float kyc = ceilf(ky), kyf = floorf(ky);
    // .astype(int32) truncates toward zero, then % S with Python semantics
    int i0 = wrap197((int)(14.f * kyc + kxc));
    int i1 = wrap197((int)(14.f * kyc + kxf));
    int i2 = wrap197((int)(14.f * kyf + kxc));
    int i3 = wrap197((int)(14.f * kyf + kxf));

    const float4* __restrict__ qr = (const float4*)(q + (size_t)t * DD);
    const size_t rb = (size_t)b * SEQ * DD;
    const float4* __restrict__ k0p = (const float4*)(k + rb + (size_t)i0 * DD);
    const float4* __restrict__ k1p = (const float4*)(k + rb + (size_t)i1 * DD);
    const float4* __restrict__ k2p = (const float4*)(k + rb + (size_t)i2 * DD);
    const float4* __restrict__ k3p = (const float4*)(k + rb + (size_t)i3 * DD);

    float s0 = 0.f, s1 = 0.f, s2 = 0.f, s3 = 0.f;
#pragma unroll
    for (int j = 0; j < 6; ++j) {
        const int e = lane + 32 * j;
        float4 qv = qr[e];
        float4 c0 = k0p[e], c1 = k1p[e], c2 = k2p[e], c3 = k3p[e];
        s0 += qv.x * c0.x + qv.y * c0.y + qv.z * c0.z + qv.w * c0.w;
        s1 += qv.x * c1.x + qv.y * c1.y + qv.z * c1.z + qv.w * c1.w;
        s2 += qv.x * c2.x + qv.y * c2.y + qv.z * c2.z + qv.w * c2.w;
        s3 += qv.x * c3.x + qv.y * c3.y + qv.z * c3.z + qv.w * c3.w;
    }
    // Butterfly reduction: every lane ends with the full dot products.
#pragma unroll
    for (int off = 16; off >= 1; off >>= 1) {
        s0 += __shfl_xor(s0, off, 32);
        s1 += __shfl_xor(s1, off, 32);
        s2 += __shfl_xor(s2, off, 32);
        s3 += __shfl_xor(s3, off, 32);
    }

    float mx = fmaxf(fmaxf(s0, s1), fmaxf(s2, s3));
    float e0 = expf(s0 - mx), e1 = expf(s1 - mx);
    float e2 = expf(s2 - mx), e3 = expf(s3 - mx);
    float inv = 1.f / (e0 + e1 + e2 + e3);
    float w0 = e0 * inv, w1 = e1 * inv, w2 = e2 * inv, w3 = e3 * inv;

    const float4* __restrict__ v0p = (const float4*)(v + rb + (size_t)i0 * DD);
    const float4* __restrict__ v1p = (const float4*)(v + rb + (size_t)i1 * DD);
    const float4* __restrict__ v2p = (const float4*)(v + rb + (size_t)i2 * DD);
    const float4* __restrict__ v3p = (const float4*)(v + rb + (size_t)i3 * DD);
#pragma unroll
    for (int j = 0; j < 6; ++j) {
        const int e = lane + 32 * j;
        float4 a = v0p[e], bb = v1p[e], c = v2p[e], d = v3p[e];
        float4 o;
        o.x = w0 * a.x + w1 * bb.x + w2 * c.x + w3 * d.x;
        o.y = w0 * a.y + w1 * bb.y + w2 * c.y + w3 * d.y;
        o.z = w0 * a.z + w1 * bb.z + w2 * c.z + w3 * d.z;
        o.w = w0 * a.w + w1 * bb.w + w2 * c.w + w3 * d.w;
        *(float4*)(orow + e * 4) = o;
    }
}

// ---------------------------------------------------------------------------
extern "C" void kernel_launch(void* const* d_in, const int* in_sizes, int n_in,
                              void* d_out, int out_size, void* d_ws, size_t ws_size,
                              hipStream_t stream) {
    const float* x        = (const float*)d_in[0];
    const float* norm_x   = (const float*)d_in[1];
    const float* norm_y   = (const float*)d_in[2];
    const float* Wq       = (const float*)d_in[3];
    const float* bq       = (const float*)d_in[4];
    const float* Wk       = (const float*)d_in[5];
    const float* bk       = (const float*)d_in[6];
    const float* Wv       = (const float*)d_in[7];
    const float* bv       = (const float*)d_in[8];
    const float* avgs     = (const float*)d_in[9];
    const float* std_devs = (const float*)d_in[10];
    const int*   img_ids  = (const int*)d_in[11];
    (void)in_sizes; (void)n_in; (void)out_size; (void)ws_size;

    float* ws  = (float*)d_ws;   // q | k | v, each 12608*768 floats
    float* out = (float*)d_out;

    dim3 g1(MTOT / 64, DD / 256, 3);   // (197, 3, 3)
    qkv_gemm<<<g1, 256, 0, stream>>>(x, Wq, bq, Wk, bk, Wv, bv, ws);

    dim3 g2((MTOT + 7) / 8);           // 1576 blocks, 8 tokens (waves) each
    gauss_attn<<<g2, 256, 0, stream>>>(ws, norm_x, norm_y, avgs, std_devs,
                                       img_ids, out);
}